// GATSage_5308579578322
// MI455X (gfx1250) — compile-verified
//
#include <hip/hip_runtime.h>
#include <hip/hip_bf16.h>

#define N0v 400000
#define N1v 75000
#define N2v 15000
#define E1v 1200000
#define E2v 240000
#define FINv 64
#define NCv 40
#define NEG 0.2f

typedef __attribute__((ext_vector_type(2))) float v2f;
typedef __attribute__((ext_vector_type(8))) float v8f;

__device__ __forceinline__ float lrelu(float x) { return x > 0.f ? x : NEG * x; }

// Sign-aware float atomic max via integer ordering trick.
__device__ __forceinline__ void atomicMaxF(float* a, float v) {
    if (v >= 0.f) atomicMax((int*)a, __float_as_int(v));
    else          atomicMin((unsigned int*)a, __float_as_uint(v));
}

// ---------------------------------------------------------------------------
// WMMA GEMM: Y[M,16] = X[M,64] @ W[64,16], fp32 via V_WMMA_F32_16X16X4_F32.
// One wave handles a 16-row tile; 16 WMMA ops sweep K=64 in steps of 4.
// ---------------------------------------------------------------------------
__global__ void k_gemm64x16_wmma(const float* __restrict__ X,
                                 const float* __restrict__ W,
                                 float* __restrict__ Y, int M) {
    int lane = threadIdx.x & 31;
    int wave = (blockIdx.x * blockDim.x + threadIdx.x) >> 5;
    int row0 = wave * 16;
    if (row0 >= M) return;
    int half = lane >> 4;    // 0: lanes 0-15, 1: lanes 16-31
    int mrow = lane & 15;    // A row / B col for this lane
    const float* xrow = X + (size_t)(row0 + mrow) * FINv;
    v8f c = {};
#pragma unroll
    for (int kk = 0; kk < 16; ++kk) {
        int kb = kk * 4 + half * 2;
        v2f a, b;
        a.x = xrow[kb];
        a.y = xrow[kb + 1];
        b.x = W[kb * 16 + mrow];
        b.y = W[(kb + 1) * 16 + mrow];
        c = __builtin_amdgcn_wmma_f32_16x16x4_f32(false, a, false, b,
                                                  (short)0, c, false, false);
    }
#pragma unroll
    for (int r = 0; r < 8; ++r) {
        int row = row0 + r + half * 8;   // C layout: VGPR r -> M=r (lo) / M=r+8 (hi)
        Y[(size_t)row * 16 + mrow] = c[r];
    }
}

// Fold W_d with att_d:  vd1[k,h] = sum_c W1d[k, h*8+c] * att1_d[h,c]  (64x2)
//                       vd2[k]   = sum_c W2d[k, c]     * att2_d[c]    (16x1)
__global__ void k_fold(const float* __restrict__ W1d, const float* __restrict__ att1d,
                       const float* __restrict__ W2d, const float* __restrict__ att2d,
                       float* __restrict__ vd1, float* __restrict__ vd2) {
    int t = threadIdx.x;
    if (t < 128) {            // 64 k-values x 2 heads
        int k = t >> 1, h = t & 1;
        float s = 0.f;
        for (int cc = 0; cc < 8; ++cc) s += W1d[k * 16 + h * 8 + cc] * att1d[h * 8 + cc];
        vd1[k * 2 + h] = s;
    }
    if (t < 16) {
        float s = 0.f;
        for (int cc = 0; cc < NCv; ++cc) s += W2d[t * NCv + cc] * att2d[cc];
        vd2[t] = s;
    }
}

// a_s per (node, head): dot of hs channel block with att_s
template <int H, int C>
__global__ void k_as(const float* __restrict__ hs, const float* __restrict__ atts,
                     float* __restrict__ as_, int N) {
    int t = blockIdx.x * blockDim.x + threadIdx.x;
    if (t >= N * H) return;
    int node = t / H, h = t % H;
    float s = 0.f;
#pragma unroll
    for (int cc = 0; cc < C; ++cc)
        s += hs[(size_t)node * (H * C) + h * C + cc] * atts[h * C + cc];
    as_[t] = s;
}

// a_d per dst node: GEMV against the folded vector
template <int H, int K>
__global__ void k_ad(const float* __restrict__ Xd, const float* __restrict__ vd,
                     float* __restrict__ ad, int Nd) {
    int i = blockIdx.x * blockDim.x + threadIdx.x;
    if (i >= Nd) return;
    float acc[H];
#pragma unroll
    for (int h = 0; h < H; ++h) acc[h] = 0.f;
    const float* xr = Xd + (size_t)i * K;
#pragma unroll 4
    for (int k = 0; k < K; ++k)
#pragma unroll
        for (int h = 0; h < H; ++h) acc[h] += xr[k] * vd[k * H + h];
#pragma unroll
    for (int h = 0; h < H; ++h) ad[i * H + h] = acc[h];
}

// Seed m[dst] with the self-loop logit (every dst gets a self loop).
template <int H>
__global__ void k_init_m(const float* __restrict__ as_, const float* __restrict__ ad,
                         float* __restrict__ m, int Nd) {
    int t = blockIdx.x * blockDim.x + threadIdx.x;
    if (t >= Nd * H) return;
    m[t] = lrelu(as_[t] + ad[t]);
}

// Pass 1 over edges: segment max of leaky-relu logits (skip src==dst).
template <int H>
__global__ void k_edge_max(const int* __restrict__ src, const int* __restrict__ dst,
                           const float* __restrict__ as_, const float* __restrict__ ad,
                           float* __restrict__ m, int E) {
    int e = blockIdx.x * blockDim.x + threadIdx.x;
    if (e >= E) return;
    int s = src[e], d = dst[e];
    if (s == d) return;
#pragma unroll
    for (int h = 0; h < H; ++h)
        atomicMaxF(&m[d * H + h], lrelu(as_[s * H + h] + ad[d * H + h]));
}

// Initialize denom/acc with the self-loop contribution (also zero-inits them).
template <int H, int C>
__global__ void k_self_init(const float* __restrict__ as_, const float* __restrict__ ad,
                            const float* __restrict__ m, const float* __restrict__ hs,
                            float* __restrict__ den, float* __restrict__ acc, int Nd) {
    int t = blockIdx.x * blockDim.x + threadIdx.x;
    if (t >= Nd * H) return;
    int i = t / H, h = t % H;
    float w = __expf(lrelu(as_[t] + ad[t]) - m[t]);
    den[t] = w;
#pragma unroll
    for (int cc = 0; cc < C; ++cc)
        acc[(size_t)i * (H * C) + h * C + cc] = w * hs[(size_t)i * (H * C) + h * C + cc];
}

// Pass 2 over edges: accumulate unnormalized w*hs[src] and w into denom.
template <int H, int C>
__global__ void k_edge_acc(const int* __restrict__ src, const int* __restrict__ dst,
                           const float* __restrict__ as_, const float* __restrict__ ad,
                           const float* __restrict__ m, const float* __restrict__ hs,
                           float* __restrict__ den, float* __restrict__ acc, int E) {
    int t = blockIdx.x * blockDim.x + threadIdx.x;
    if (t >= E * H) return;
    int e = t / H, h = t % H;
    int s = src[e], d = dst[e];
    if (s == d) return;
    float w = __expf(lrelu(as_[s * H + h] + ad[d * H + h]) - m[d * H + h]);
    atomicAdd(&den[d * H + h], w);
    const float* hrow = hs + (size_t)s * (H * C) + h * C;
    float* arow = acc + (size_t)d * (H * C) + h * C;
#pragma unroll
    for (int cc = 0; cc < C; ++cc) atomicAdd(&arow[cc], w * hrow[cc]);
}

// Layer-1 epilogue: normalize, add bias, ELU.
__global__ void k_final1(const float* __restrict__ acc, const float* __restrict__ den,
                         const float* __restrict__ b1, float* __restrict__ h1, int Nd) {
    int t = blockIdx.x * blockDim.x + threadIdx.x;
    if (t >= Nd * 16) return;
    int i = t >> 4, cc = t & 15;
    float v = acc[t] / den[i * 2 + (cc >> 3)] + b1[cc];
    h1[t] = v > 0.f ? v : (__expf(v) - 1.f);
}

// Layer-2 source transform: hs2[N1,40] = h1[N1,16] @ W2s[16,40] (tiny; VALU).
__global__ void k_gemm2(const float* __restrict__ h1, const float* __restrict__ W2s,
                        float* __restrict__ hs2, int N) {
    int t = blockIdx.x * blockDim.x + threadIdx.x;
    if (t >= N * NCv) return;
    int i = t / NCv, cc = t % NCv;
    const float* hr = h1 + (size_t)i * 16;
    float s = 0.f;
#pragma unroll
    for (int k = 0; k < 16; ++k) s += hr[k] * W2s[k * NCv + cc];
    hs2[t] = s;
}

// Layer-2 epilogue: normalize, bias, log_softmax over 40 classes.
__global__ void k_final2(const float* __restrict__ acc, const float* __restrict__ den,
                         const float* __restrict__ b2, float* __restrict__ out, int Nd) {
    int i = blockIdx.x * blockDim.x + threadIdx.x;
    if (i >= Nd) return;
    float inv = 1.f / den[i];
    float v[NCv];
    float mx = -3.4e38f;
#pragma unroll
    for (int cc = 0; cc < NCv; ++cc) {
        v[cc] = acc[(size_t)i * NCv + cc] * inv + b2[cc];
        mx = fmaxf(mx, v[cc]);
    }
    float se = 0.f;
#pragma unroll
    for (int cc = 0; cc < NCv; ++cc) se += __expf(v[cc] - mx);
    float lse = __logf(se) + mx;
#pragma unroll
    for (int cc = 0; cc < NCv; ++cc) out[(size_t)i * NCv + cc] = v[cc] - lse;
}

static inline int ceil_div(int a, int b) { return (a + b - 1) / b; }

extern "C" void kernel_launch(void* const* d_in, const int* in_sizes, int n_in,
                              void* d_out, int out_size, void* d_ws, size_t ws_size,
                              hipStream_t stream) {
    const float* x     = (const float*)d_in[0];
    const int*  e1s    = (const int*)d_in[1];
    const int*  e1d    = (const int*)d_in[2];
    const int*  e2s    = (const int*)d_in[3];
    const int*  e2d    = (const int*)d_in[4];
    const float* W1s   = (const float*)d_in[5];
    const float* W1d   = (const float*)d_in[6];
    const float* att1s = (const float*)d_in[7];
    const float* att1d = (const float*)d_in[8];
    const float* b1    = (const float*)d_in[9];
    const float* W2s   = (const float*)d_in[10];
    const float* W2d   = (const float*)d_in[11];
    const float* att2s = (const float*)d_in[12];
    const float* att2d = (const float*)d_in[13];
    const float* b2    = (const float*)d_in[14];
    float* out = (float*)d_out;

    // Carve workspace (floats)
    float* ws = (float*)d_ws;
    size_t o = 0;
    float* hs1  = ws + o; o += (size_t)N0v * 16;
    float* as1  = ws + o; o += (size_t)N0v * 2;
    float* ad1  = ws + o; o += (size_t)N1v * 2;
    float* m1   = ws + o; o += (size_t)N1v * 2;
    float* den1 = ws + o; o += (size_t)N1v * 2;
    float* acc1 = ws + o; o += (size_t)N1v * 16;
    float* h1   = ws + o; o += (size_t)N1v * 16;
    float* hs2  = ws + o; o += (size_t)N1v * NCv;
    float* as2  = ws + o; o += (size_t)N1v;
    float* ad2  = ws + o; o += (size_t)N2v;
    float* m2   = ws + o; o += (size_t)N2v;
    float* den2 = ws + o; o += (size_t)N2v;
    float* acc2 = ws + o; o += (size_t)N2v * NCv;
    float* vd1  = ws + o; o += 128;
    float* vd2  = ws + o; o += 16;
    (void)ws_size; (void)in_sizes; (void)n_in; (void)out_size;

    const int B = 256;

    // Fold dest-transforms into GEMV vectors
    k_fold<<<1, 128, 0, stream>>>(W1d, att1d, W2d, att2d, vd1, vd2);

    // ---- Layer 1 ----
    // hs1 = x @ W1s  (WMMA; 400000/16 = 25000 waves, 8 waves/block)
    k_gemm64x16_wmma<<<ceil_div(N0v / 16, 8), B, 0, stream>>>(x, W1s, hs1, N0v);
    k_as<2, 8><<<ceil_div(N0v * 2, B), B, 0, stream>>>(hs1, att1s, as1, N0v);
    k_ad<2, 64><<<ceil_div(N1v, B), B, 0, stream>>>(x, vd1, ad1, N1v);

    k_init_m<2><<<ceil_div(N1v * 2, B), B, 0, stream>>>(as1, ad1, m1, N1v);
    k_edge_max<2><<<ceil_div(E1v, B), B, 0, stream>>>(e1s, e1d, as1, ad1, m1, E1v);
    k_self_init<2, 8><<<ceil_div(N1v * 2, B), B, 0, stream>>>(as1, ad1, m1, hs1, den1, acc1, N1v);
    k_edge_acc<2, 8><<<ceil_div(E1v * 2, B), B, 0, stream>>>(e1s, e1d, as1, ad1, m1, hs1,
                                                             den1, acc1, E1v);
    k_final1<<<ceil_div(N1v * 16, B), B, 0, stream>>>(acc1, den1, b1, h1, N1v);

    // ---- Layer 2 ----
    k_gemm2<<<ceil_div(N1v * NCv, B), B, 0, stream>>>(h1, W2s, hs2, N1v);
    k_as<1, NCv><<<ceil_div(N1v, B), B, 0, stream>>>(hs2, att2s, as2, N1v);
    k_ad<1, 16><<<ceil_div(N2v, B), B, 0, stream>>>(h1, vd2, ad2, N2v);

    k_init_m<1><<<ceil_div(N2v, B), B, 0, stream>>>(as2, ad2, m2, N2v);
    k_edge_max<1><<<ceil_div(E2v, B), B, 0, stream>>>(e2s, e2d, as2, ad2, m2, E2v);
    k_self_init<1, NCv><<<ceil_div(N2v, B), B, 0, stream>>>(as2, ad2, m2, hs2, den2, acc2, N2v);
    k_edge_acc<1, NCv><<<ceil_div(E2v, B), B, 0, stream>>>(e2s, e2d, as2, ad2, m2, hs2,
                                                           den2, acc2, E2v);
    k_final2<<<ceil_div(N2v, B), B, 0, stream>>>(acc2, den2, b2, out, N2v);
}